// CustomModel_30176440221957
// MI455X (gfx1250) — compile-verified
//
#include <hip/hip_runtime.h>
#include <cstdint>
#include <cstddef>

// Problem constants from the reference (B,T,V,L fixed by setup_inputs()).
#define B_    64
#define T_    2000
#define V_    128
#define L_    200
#define S_    401          // 2*L + 1
#define NPL   13           // states per lane: 13*32 = 416 >= 401
#define RING  32           // LDS ring slots (rows of V_ floats) -> 16 KB
#define DEPTH 24           // async prefetch depth (covers HBM latency)
#define NEGF  (-1.0e9f)    // log-space "zero", matches reference

// ---------- wave32 helpers (raw bpermute; wave32 lane id == threadIdx.x) ----

static __device__ __forceinline__ float bitf(int i)  { return __builtin_bit_cast(float, i); }
static __device__ __forceinline__ int   biti(float f){ return __builtin_bit_cast(int, f); }

static __device__ __forceinline__ float lane_xor_f(float v, int mask, int lane) {
  int idx = ((lane ^ mask) & 31) << 2;                 // byte index for bpermute
  return bitf(__builtin_amdgcn_ds_bpermute(idx, biti(v)));
}
static __device__ __forceinline__ float wave_max(float v, int lane) {
  #pragma unroll
  for (int m = 16; m >= 1; m >>= 1) v = fmaxf(v, lane_xor_f(v, m, lane));
  return v;
}
static __device__ __forceinline__ float wave_sum(float v, int lane) {
  #pragma unroll
  for (int m = 16; m >= 1; m >>= 1) v += lane_xor_f(v, m, lane);
  return v;
}
// value of (lane-1); NEGF for lane 0
static __device__ __forceinline__ float lane_up1_f(float v, int lane) {
  int idx = ((lane - 1) & 31) << 2;
  float r = bitf(__builtin_amdgcn_ds_bpermute(idx, biti(v)));
  return (lane == 0) ? NEGF : r;
}

// ---------- CDNA5 async global->LDS path --------------------------------

// Low 32 bits of a flat pointer into LDS are the wave-relative LDS offset
// (aperture bits live in [63:32] per CDNA5 ISA §10.2).
static __device__ __forceinline__ uint32_t lds_off(const void* p) {
  return (uint32_t)(uintptr_t)p;
}

// One row per wave: each of the 32 lanes moves 16B -> 512B = V_ floats.
static __device__ __forceinline__ void async_row_load(uint32_t ldsoff, const float* g) {
  asm volatile("global_load_async_to_lds_b128 %0, %1, off"
               :: "v"(ldsoff), "v"(g)
               : "memory");
}
static __device__ __forceinline__ void wait_async_steady() {   // <= DEPTH-1
  asm volatile("s_wait_asynccnt 23" ::: "memory");
}
static __device__ __forceinline__ void wait_async_all() {
  asm volatile("s_wait_asynccnt 0" ::: "memory");
}

// 2-exp logaddexp3: one of the three terms is always exp(0)=1.
static __device__ __forceinline__ float logaddexp3(float x, float y, float z) {
  float lo  = fminf(x, y);
  float hi  = fmaxf(x, y);
  float m3  = fmaxf(hi, z);
  float mid = fminf(hi, z);
  float ss  = 1.0f + __expf(lo - m3) + __expf(mid - m3);
  return m3 + __logf(ss);
}

// ---------- main CTC forward kernel: one wave32 per batch element -------

__global__ __launch_bounds__(32)
void ctc_forward_kernel(const float* __restrict__ logits,  // [B,T,V]
                        const int*   __restrict__ labels,  // [B,L]
                        float*       __restrict__ nll)     // [B] workspace
{
  __shared__ __align__(16) float rowbuf[RING][V_];   // 16 KB prefetch ring
  __shared__ int extbuf[NPL * 32];                   // 416 padded ext labels

  const int b    = (int)blockIdx.x;
  const int lane = (int)threadIdx.x;

  // Extended label sequence: blank, l1, blank, l2, ..., blank (pad = blank).
  #pragma unroll
  for (int i = 0; i < NPL; ++i) {
    int s = i * 32 + lane;
    int e = V_ - 1;
    if (s < S_ && (s & 1)) e = labels[b * L_ + (s >> 1)];
    extbuf[s] = e;
  }
  __syncthreads();

  // Hoist per-lane ext values + skip mask into registers.
  int exts[NPL];
  unsigned skipm = 0;
  #pragma unroll
  for (int i = 0; i < NPL; ++i) {
    int s = lane * NPL + i;
    int e = extbuf[s];
    exts[i] = e;
    if (s >= 2 && s < S_ && e != extbuf[s - 2]) skipm |= (1u << i);
  }
  __syncthreads();

  const float* base = logits + (size_t)b * (size_t)T_ * (size_t)V_ + (size_t)(lane * 4);

  // Prologue: prefetch rows 0..DEPTH-1.
  for (int r = 0; r < DEPTH; ++r) {
    async_row_load(lds_off(&rowbuf[r & (RING - 1)][lane * 4]),
                   base + (size_t)r * V_);
  }

  // Softmax + gather for one row -> lpv (log-probs of my 13 ext labels).
  // (Macro-style lambda keeps it in one place; compiler inlines.)
  auto softmax_gather = [&](int t, float* lpv) {
    const float* row = rowbuf[t & (RING - 1)];
    float4 x = *(const float4*)(row + lane * 4);
    float m = fmaxf(fmaxf(x.x, x.y), fmaxf(x.z, x.w));
    m = wave_max(m, lane);
    float es = __expf(x.x - m) + __expf(x.y - m) + __expf(x.z - m) + __expf(x.w - m);
    es = wave_sum(es, lane);
    const float logZ = m + __logf(es);
    #pragma unroll
    for (int i = 0; i < NPL; ++i) lpv[i] = row[exts[i]] - logZ;
  };

  // Prologue compute: lpv for row 0.
  wait_async_steady();            // rows 0..23 outstanding -> row 0 done
  float lpv[NPL];
  softmax_gather(0, lpv);

  float a[NPL];
  #pragma unroll
  for (int i = 0; i < NPL; ++i) a[i] = NEGF;
  float c1 = NEGF, c2 = NEGF;     // old alpha[s-1], alpha[s-2] from prev lane

  for (int t = 0; t < T_; ++t) {
    const bool deep = (t + DEPTH < T_);
    // Keep the ring full: issue row t+DEPTH (slot == (t-8)&31, long dead).
    if (deep) {
      async_row_load(lds_off(&rowbuf[(t + DEPTH) & (RING - 1)][lane * 4]),
                     base + (size_t)(t + DEPTH) * V_);
    }

    // ---- alpha update for step t (independent of row t+1 pipeline) ----
    if (t == 0) {
      #pragma unroll
      for (int i = 0; i < NPL; ++i) {
        int s = lane * NPL + i;
        a[i] = (s < 2) ? lpv[i] : NEGF;
      }
    } else {
      float k1 = c1, k2 = c2;
      #pragma unroll
      for (int i = 0; i < NPL; ++i) {
        float ai = a[i];
        float a2 = ((skipm >> i) & 1u) ? k2 : NEGF;
        a[i] = logaddexp3(ai, k1, a2) + lpv[i];
        k2 = k1; k1 = ai;
      }
    }

    // ---- pipeline: softmax + gather for row t+1 (hides DS-chain stalls
    //      behind the independent logaddexp work above) ----
    if (t + 1 < T_) {
      if (deep) wait_async_steady();   // rows t+1..t+24 outstanding
      else      wait_async_all();      // tail: wait everything
      softmax_gather(t + 1, lpv);
    }

    // ---- boundary shuffles for the next step (latency hidden by the
    //      next iteration's softmax block) ----
    c1 = lane_up1_f(a[NPL - 1], lane);
    c2 = lane_up1_f(a[NPL - 2], lane);
  }

  // ll = logaddexp(alpha[S-1], alpha[S-2]); s=400 -> lane30,i=10; s=399 -> lane30,i=9.
  if (lane == 30) {
    float x0 = a[9], x1 = a[10];
    float mm = fmaxf(x0, x1);
    float ll = mm + __logf(__expf(x0 - mm) + __expf(x1 - mm));
    nll[b] = -ll;
  }
}

// ---------- final mean over B=64 per-example losses ----------------------

__global__ __launch_bounds__(32)
void ctc_mean_kernel(const float* __restrict__ nll, float* __restrict__ out)
{
  int lane = (int)threadIdx.x;
  float v = nll[lane] + nll[lane + 32];
  v = wave_sum(v, lane);
  if (lane == 0) out[0] = v * (1.0f / (float)B_);
}

// ---------- launch -------------------------------------------------------

extern "C" void kernel_launch(void* const* d_in, const int* in_sizes, int n_in,
                              void* d_out, int out_size, void* d_ws, size_t ws_size,
                              hipStream_t stream)
{
  (void)in_sizes; (void)n_in; (void)out_size; (void)ws_size;
  const float* logits = (const float*)d_in[0];   // [B,T,V] float32
  const int*   labels = (const int*)d_in[1];     // [B,L] int
  float* per_b = (float*)d_ws;                   // B floats of scratch
  float* out   = (float*)d_out;                  // scalar

  ctc_forward_kernel<<<B_, 32, 0, stream>>>(logits, labels, per_b);
  ctc_mean_kernel<<<1, 32, 0, stream>>>(per_b, out);
}